// LightweightSelfAttention_90529320665265
// MI455X (gfx1250) — compile-verified
//
#include <hip/hip_runtime.h>
#include <math.h>

typedef __attribute__((ext_vector_type(2))) float v2f;
typedef __attribute__((ext_vector_type(8))) float v8f;
typedef __attribute__((ext_vector_type(4))) int v4i;

#define DIMD   512
#define S_LEN  2048
#define BATCH  2
#define NHEAD  16
#define HDIM   32
#define WIN    64
#define MTOT   (BATCH * S_LEN)          /* 4096 rows */
#define QSCALE 0.17677669529663687f     /* 32^-0.5 */

#define AS1 __attribute__((address_space(1)))
#define AS3 __attribute__((address_space(3)))

#if __has_builtin(__builtin_amdgcn_global_load_async_to_lds_b128) && \
    __has_builtin(__builtin_amdgcn_s_wait_asynccnt)
#define HAS_ASYNC_LDS 1
#else
#define HAS_ASYNC_LDS 0
#endif

__device__ __forceinline__ void copy16_to_lds(const float* g, float* l) {
#if HAS_ASYNC_LDS
  // async global -> LDS, tracked by ASYNCcnt (no VGPR staging)
  __builtin_amdgcn_global_load_async_to_lds_b128(
      (AS1 v4i*)(v4i*)const_cast<float*>(g), (AS3 v4i*)(v4i*)l, 0, 0);
#else
  *(float4*)l = *(const float4*)g;
#endif
}

template <int N>
__device__ __forceinline__ void async_wait() {
#if HAS_ASYNC_LDS
  __builtin_amdgcn_s_wait_asynccnt(N);
#endif
}

__device__ __forceinline__ v8f zero8() {
  v8f z;
#pragma unroll
  for (int i = 0; i < 8; ++i) z[i] = 0.0f;
  return z;
}

__device__ __forceinline__ v8f wmma4(v2f a, v2f b, v8f c) {
  // D = A(16x4 f32) * B(4x16 f32) + C(16x16 f32), full fp32 precision
  return __builtin_amdgcn_wmma_f32_16x16x4_f32(false, a, false, b, (short)0, c,
                                               false, false);
}

// ---------------------------------------------------------------------------
// Kernel 1: qkv = x @ W_qkv + b ; q = silu(qk)*scale, k = tanh(qk), v = rest
// M=4096, N=1024, K=512. Block tile 64(M) x 128(N), KC=32, double-buffered
// async global->LDS tile loads. 256 thr = 8 waves, wave computes 16x64.
// ---------------------------------------------------------------------------
__global__ __launch_bounds__(256) void qkv_kernel(
    const float* __restrict__ x, const float* __restrict__ Wq,
    const float* __restrict__ bq, float* __restrict__ qout,
    float* __restrict__ kout, float* __restrict__ vout) {
  __shared__ float Xs[2][64 * 32];
  __shared__ float Ws[2][32 * 128];
  const int t = threadIdx.x;
  const int lane = t & 31, w = t >> 5;
  const int lo = lane & 15, hi = lane >> 4;
  const int wm = w & 3, wn = w >> 2;
  const int blockM = blockIdx.y * 64;
  const int blockN = blockIdx.x * 128;

  v8f acc[4];
#pragma unroll
  for (int i = 0; i < 4; ++i) acc[i] = zero8();

  const int xi = t >> 2;        // 0..63 (row of X tile)
  const int xk = (t & 3) * 8;   // 0,8,16,24
  const int wr = t >> 3;        // 0..31 (row of W tile)
  const int wc = (t & 7) * 16;  // col within 128

  auto load_chunk = [&](int k0, int buf) {
    const float* xg = &x[(size_t)(blockM + xi) * DIMD + k0 + xk];
    float* xl = &Xs[buf][xi * 32 + xk];
    copy16_to_lds(xg, xl);
    copy16_to_lds(xg + 4, xl + 4);
    const float* wg = &Wq[(size_t)(k0 + wr) * (2 * DIMD) + blockN + wc];
    float* wl = &Ws[buf][wr * 128 + wc];
    copy16_to_lds(wg, wl);
    copy16_to_lds(wg + 4, wl + 4);
    copy16_to_lds(wg + 8, wl + 8);
    copy16_to_lds(wg + 12, wl + 12);
  };

  load_chunk(0, 0);
  int buf = 0;
  for (int k0 = 0; k0 < DIMD; k0 += 32, buf ^= 1) {
    const bool more = (k0 + 32 < DIMD);
    if (more) {
      load_chunk(k0 + 32, buf ^ 1);  // prefetch next tile (async)
      async_wait<6>();               // current tile has landed (in-order)
    } else {
      async_wait<0>();
    }
    __syncthreads();
#pragma unroll
    for (int kb = 0; kb < 8; ++kb) {
      const int kk = kb * 4 + hi * 2;
      v2f a = *(const v2f*)&Xs[buf][(wm * 16 + lo) * 32 + kk];
#pragma unroll
      for (int nt = 0; nt < 4; ++nt) {
        v2f b;
        b.x = Ws[buf][kk * 128 + wn * 64 + nt * 16 + lo];
        b.y = Ws[buf][(kk + 1) * 128 + wn * 64 + nt * 16 + lo];
        acc[nt] = wmma4(a, b, acc[nt]);
      }
    }
    __syncthreads();
  }

#pragma unroll
  for (int nt = 0; nt < 4; ++nt) {
    const int gn = blockN + wn * 64 + nt * 16 + lo;
    const float bias = bq[gn];
#pragma unroll
    for (int r = 0; r < 8; ++r) {
      const int gm = blockM + wm * 16 + r + 8 * hi;
      const float val = acc[nt][r] + bias;
      if (gn < DIMD) {
        const float sig = 1.0f / (1.0f + __expf(-val));
        qout[(size_t)gm * DIMD + gn] = val * sig * QSCALE;
        kout[(size_t)gm * DIMD + gn] = tanhf(val);
      } else {
        vout[(size_t)gm * DIMD + (gn - DIMD)] = val;
      }
    }
  }
}

// ---------------------------------------------------------------------------
// Kernel 2: per-(batch,head) column sums of V (for out-of-band correction)
// ---------------------------------------------------------------------------
__global__ __launch_bounds__(256) void vsum_kernel(const float* __restrict__ vb,
                                                   float* __restrict__ vs) {
  __shared__ float red[8][32];
  const int bh = blockIdx.x;  // b*16 + h
  const int b = bh >> 4, h = bh & 15;
  const int t = threadIdx.x;
  const int d = t & 31, c = t >> 5;
  float s = 0.0f;
  for (int j = c; j < S_LEN; j += 8)
    s += vb[(size_t)(b * S_LEN + j) * DIMD + h * HDIM + d];
  red[c][d] = s;
  __syncthreads();
  if (c == 0) {
    float tot = 0.0f;
#pragma unroll
    for (int i = 0; i < 8; ++i) tot += red[i][d];
    vs[bh * HDIM + d] = tot;
  }
}

// ---------------------------------------------------------------------------
// Kernel 3: banded attention with global correction.
// One wave per (16-row query tile, head, batch).
// scores = Q Kt (WMMA), p = inband ? exp(s) : 1 ; PV via WMMA through LDS.
// out = (PV_tiles + Vtotal - Vtiles) / (Z_tiles + (S - Ntiles))
// ---------------------------------------------------------------------------
__global__ __launch_bounds__(32) void attn_kernel(
    const float* __restrict__ qb, const float* __restrict__ kmat,
    const float* __restrict__ vb, const float* __restrict__ vs,
    float* __restrict__ ob) {
  __shared__ float pl[16 * 16];
  const int lane = threadIdx.x;
  const int lo = lane & 15, hi = lane >> 4;
  const int i0 = blockIdx.x * 16;
  const int h = blockIdx.y;
  const int b = blockIdx.z;
  const size_t base = (size_t)(b * S_LEN) * DIMD + h * HDIM;

  // Q rows in WMMA A layout: lane holds M=lo, K = kb*4 + hi*2 + {0,1}
  v2f qa[8];
#pragma unroll
  for (int kbi = 0; kbi < 8; ++kbi)
    qa[kbi] = *(const v2f*)&qb[base + (size_t)(i0 + lo) * DIMD + kbi * 4 + hi * 2];

  const int jlo = (i0 - WIN > 0) ? (i0 - WIN) : 0;          // multiple of 16
  const int jhiP1 = (i0 + 16 + WIN < S_LEN) ? (i0 + 16 + WIN) : S_LEN;
  const int NC = jhiP1 - jlo;

  // per-dim V sum over the covered tiles (lane == dim 0..31)
  float vt = 0.0f;
  for (int j = jlo; j < jhiP1; ++j) vt += vb[base + (size_t)j * DIMD + lane];

  v8f acc0 = zero8(), acc1 = zero8();
  float zs[8];
#pragma unroll
  for (int r = 0; r < 8; ++r) zs[r] = 0.0f;

  for (int j0 = jlo; j0 < jhiP1; j0 += 16) {
    // S_tile = Q(16x32) @ K_tile^T(32x16)
    v8f sc = zero8();
#pragma unroll
    for (int kbi = 0; kbi < 8; ++kbi) {
      v2f bb = *(const v2f*)&kmat[base + (size_t)(j0 + lo) * DIMD + kbi * 4 + hi * 2];
      sc = wmma4(qa[kbi], bb, sc);
    }
    // mask + exp in C layout: (M = r + 8*hi, N = lo)
#pragma unroll
    for (int r = 0; r < 8; ++r) {
      const int i = i0 + r + 8 * hi;
      const int j = j0 + lo;
      int dlt = i - j;
      if (dlt < 0) dlt = -dlt;
      const float p = (dlt <= WIN) ? __expf(sc[r]) : 1.0f;
      zs[r] += p;
      pl[(r + 8 * hi) * 16 + lo] = p;
    }
    __syncthreads();
    // out_tile += P(16x16) @ V_tile(16x32), split into two N halves
#pragma unroll
    for (int kbi = 0; kbi < 4; ++kbi) {
      const int kk = kbi * 4 + hi * 2;
      v2f pa = *(const v2f*)&pl[lo * 16 + kk];
      v2f b0, b1;
      b0.x = vb[base + (size_t)(j0 + kk) * DIMD + lo];
      b0.y = vb[base + (size_t)(j0 + kk + 1) * DIMD + lo];
      b1.x = vb[base + (size_t)(j0 + kk) * DIMD + 16 + lo];
      b1.y = vb[base + (size_t)(j0 + kk + 1) * DIMD + 16 + lo];
      acc0 = wmma4(pa, b0, acc0);
      acc1 = wmma4(pa, b1, acc1);
    }
    __syncthreads();
  }

  const float vt0 = __shfl(vt, lo, 32);
  const float vt1 = __shfl(vt, 16 + lo, 32);
  const float vs0 = vs[(b * NHEAD + h) * HDIM + lo];
  const float vs1 = vs[(b * NHEAD + h) * HDIM + 16 + lo];
  const float zoff = (float)(S_LEN - NC);

#pragma unroll
  for (int r = 0; r < 8; ++r) {
    float z = zs[r];
#pragma unroll
    for (int m = 1; m < 16; m <<= 1) z += __shfl_xor(z, m, 32);
    z += zoff;
    const float inv = 1.0f / z;
    const int i = i0 + r + 8 * hi;
    ob[base + (size_t)i * DIMD + lo] = (acc0[r] + vs0 - vt0) * inv;
    ob[base + (size_t)i * DIMD + 16 + lo] = (acc1[r] + vs1 - vt1) * inv;
  }
}

// ---------------------------------------------------------------------------
// Kernel 4: y = x + attn @ W_out + b_out (tiled WMMA GEMM, N=512,
// double-buffered async tile loads)
// ---------------------------------------------------------------------------
__global__ __launch_bounds__(256) void proj_kernel(
    const float* __restrict__ a, const float* __restrict__ Wo,
    const float* __restrict__ bo, const float* __restrict__ x,
    float* __restrict__ yb) {
  __shared__ float Xs[2][64 * 32];
  __shared__ float Ws[2][32 * 128];
  const int t = threadIdx.x;
  const int lane = t & 31, w = t >> 5;
  const int lo = lane & 15, hi = lane >> 4;
  const int wm = w & 3, wn = w >> 2;
  const int blockM = blockIdx.y * 64;
  const int blockN = blockIdx.x * 128;

  v8f acc[4];
#pragma unroll
  for (int i = 0; i < 4; ++i) acc[i] = zero8();

  const int xi = t >> 2;
  const int xk = (t & 3) * 8;
  const int wr = t >> 3;
  const int wc = (t & 7) * 16;

  auto load_chunk = [&](int k0, int buf) {
    const float* xg = &a[(size_t)(blockM + xi) * DIMD + k0 + xk];
    float* xl = &Xs[buf][xi * 32 + xk];
    copy16_to_lds(xg, xl);
    copy16_to_lds(xg + 4, xl + 4);
    const float* wg = &Wo[(size_t)(k0 + wr) * DIMD + blockN + wc];
    float* wl = &Ws[buf][wr * 128 + wc];
    copy16_to_lds(wg, wl);
    copy16_to_lds(wg + 4, wl + 4);
    copy16_to_lds(wg + 8, wl + 8);
    copy16_to_lds(wg + 12, wl + 12);
  };

  load_chunk(0, 0);
  int buf = 0;
  for (int k0 = 0; k0 < DIMD; k0 += 32, buf ^= 1) {
    const bool more = (k0 + 32 < DIMD);
    if (more) {
      load_chunk(k0 + 32, buf ^ 1);
      async_wait<6>();
    } else {
      async_wait<0>();
    }
    __syncthreads();
#pragma unroll
    for (int kb = 0; kb < 8; ++kb) {
      const int kk = kb * 4 + hi * 2;
      v2f av = *(const v2f*)&Xs[buf][(wm * 16 + lo) * 32 + kk];
#pragma unroll
      for (int nt = 0; nt < 4; ++nt) {
        v2f bv;
        bv.x = Ws[buf][kk * 128 + wn * 64 + nt * 16 + lo];
        bv.y = Ws[buf][(kk + 1) * 128 + wn * 64 + nt * 16 + lo];
        acc[nt] = wmma4(av, bv, acc[nt]);
      }
    }
    __syncthreads();
  }

#pragma unroll
  for (int nt = 0; nt < 4; ++nt) {
    const int gn = blockN + wn * 64 + nt * 16 + lo;
    const float bias = bo[gn];
#pragma unroll
    for (int r = 0; r < 8; ++r) {
      const int gm = blockM + wm * 16 + r + 8 * hi;
      yb[(size_t)gm * DIMD + gn] =
          acc[nt][r] + bias + x[(size_t)gm * DIMD + gn];
    }
  }
}

// ---------------------------------------------------------------------------
// Kernel 5: LayerNorm over last dim (512), one block per row
// ---------------------------------------------------------------------------
__global__ __launch_bounds__(256) void ln_kernel(const float* __restrict__ yb,
                                                 const float* __restrict__ g,
                                                 const float* __restrict__ bta,
                                                 float* __restrict__ out) {
  __shared__ float ssum[256];
  __shared__ float ssq[256];
  const int row = blockIdx.x;
  const int t = threadIdx.x;
  const float y0 = yb[(size_t)row * DIMD + t];
  const float y1 = yb[(size_t)row * DIMD + t + 256];
  ssum[t] = y0 + y1;
  ssq[t] = y0 * y0 + y1 * y1;
  __syncthreads();
  for (int s = 128; s > 0; s >>= 1) {
    if (t < s) {
      ssum[t] += ssum[t + s];
      ssq[t] += ssq[t + s];
    }
    __syncthreads();
  }
  const float mu = ssum[0] * (1.0f / DIMD);
  const float var = ssq[0] * (1.0f / DIMD) - mu * mu;
  const float rs = rsqrtf(var + 1e-5f);
  out[(size_t)row * DIMD + t] = (y0 - mu) * rs * g[t] + bta[t];
  out[(size_t)row * DIMD + t + 256] = (y1 - mu) * rs * g[t + 256] + bta[t + 256];
}

// ---------------------------------------------------------------------------
extern "C" void kernel_launch(void* const* d_in, const int* in_sizes, int n_in,
                              void* d_out, int out_size, void* d_ws,
                              size_t ws_size, hipStream_t stream) {
  const float* x = (const float*)d_in[0];
  const float* Wq = (const float*)d_in[1];
  const float* bq = (const float*)d_in[2];
  const float* Wo = (const float*)d_in[3];
  const float* bo = (const float*)d_in[4];
  const float* gam = (const float*)d_in[5];
  const float* bet = (const float*)d_in[6];
  float* out = (float*)d_out;
  float* ws = (float*)d_ws;

  const size_t MS = (size_t)MTOT * DIMD;  // 2097152 floats
  float* qbuf = ws;
  float* kbuf = ws + MS;
  float* vbuf = ws + 2 * MS;
  float* abuf = ws + 3 * MS;
  float* vsb = ws + 4 * MS;   // B*NH*HD = 1024 floats
  float* ybuf = qbuf;         // q is dead after attention; reuse

  qkv_kernel<<<dim3(8, 64), 256, 0, stream>>>(x, Wq, bq, qbuf, kbuf, vbuf);
  vsum_kernel<<<BATCH * NHEAD, 256, 0, stream>>>(vbuf, vsb);
  attn_kernel<<<dim3(S_LEN / 16, NHEAD, BATCH), 32, 0, stream>>>(qbuf, kbuf,
                                                                 vbuf, vsb,
                                                                 abuf);
  proj_kernel<<<dim3(4, 64), 256, 0, stream>>>(abuf, Wo, bo, x, ybuf);
  ln_kernel<<<MTOT, 256, 0, stream>>>(ybuf, gam, bet, out);
}